// gcn_21766894256615
// MI455X (gfx1250) — compile-verified
//
#include <hip/hip_runtime.h>
#include <hip/hip_bf16.h>
#include <math.h>

typedef __attribute__((ext_vector_type(2))) float v2f;
typedef __attribute__((ext_vector_type(8))) float v8f;

#define NFEAT   512
#define HIDDEN  128
#define NCLS    40

// ---------------------------------------------------------------------------
// Degree: deg[dst] += 1 (float atomics; integer-valued sums < 2^24 are exact
// and order-independent -> deterministic).
__global__ void k_degree(const int* __restrict__ dst, float* __restrict__ deg, int e) {
    int t = blockIdx.x * blockDim.x + threadIdx.x;
    if (t < e) atomicAdd(&deg[dst[t]], 1.0f);
}

// deg -> 1/sqrt(deg+1) in place
__global__ void k_invsqrt(float* __restrict__ deg, int n) {
    int t = blockIdx.x * blockDim.x + threadIdx.x;
    if (t < n) deg[t] = rsqrtf(deg[t] + 1.0f);
}

// ---------------------------------------------------------------------------
// GEMM1: h1[n,128] = x[n,512] @ W1[512,128]  via V_WMMA_F32_16X16X4_F32.
// One wave computes a full 16-row x 128-col stripe (8 accumulators).
__global__ void k_gemm1(const float* __restrict__ x, const float* __restrict__ W1,
                        float* __restrict__ h1, int n) {
    int wave = (int)((blockIdx.x * blockDim.x + threadIdx.x) >> 5);
    int lane = threadIdx.x & 31;
    int m0 = wave * 16;
    if (m0 >= n) return;                       // wave-uniform: EXEC stays all-ones

    int row = lane & 15;                       // M for A-frag / N for B-frag
    int kh  = (lane >> 4) << 1;                // lanes 16-31 hold K+2,K+3

    v8f acc[8];
#pragma unroll
    for (int t = 0; t < 8; ++t) acc[t] = (v8f){0.f,0.f,0.f,0.f,0.f,0.f,0.f,0.f};

    int arow = m0 + row; if (arow >= n) arow = n - 1;      // safe clamp
    const float* ap = x + (size_t)arow * NFEAT;

    for (int k = 0; k < NFEAT; k += 4) {
        v2f a;
        a.x = ap[k + kh + 0];
        a.y = ap[k + kh + 1];
        const float* bp = W1 + (size_t)(k + kh) * HIDDEN + row;
#pragma unroll
        for (int t = 0; t < 8; ++t) {
            v2f b;
            b.x = bp[t * 16];
            b.y = bp[t * 16 + HIDDEN];
            acc[t] = __builtin_amdgcn_wmma_f32_16x16x4_f32(
                false, a, false, b, (short)0, acc[t], false, false);
        }
    }

    // D layout: VGPR r -> M = r (lanes 0-15) or M = 8+r (lanes 16-31), N = lane&15
    int mo = (lane >> 4) * 8;
    int nc = lane & 15;
#pragma unroll
    for (int t = 0; t < 8; ++t) {
#pragma unroll
        for (int r = 0; r < 8; ++r) {
            int mrow = m0 + mo + r;
            if (mrow < n) h1[(size_t)mrow * HIDDEN + t * 16 + nc] = acc[t][r];
        }
    }
}

// ---------------------------------------------------------------------------
// Edge scatter layer 1: agg[dst] += h1[src] * invsq[src]*invsq[dst]
// One wave per edge; each lane: float4 slice (4 channels) + 4 f32 atomics.
__global__ void k_edge1(const int* __restrict__ src, const int* __restrict__ dst,
                        const float* __restrict__ h1, const float* __restrict__ invsq,
                        float* __restrict__ agg, int e) {
    long long gid = (long long)blockIdx.x * blockDim.x + threadIdx.x;
    int edge = (int)(gid >> 5);
    int lane = (int)(gid & 31);
    if (edge >= e) return;
    int s = src[edge], d = dst[edge];
    float coef = invsq[s] * invsq[d];
    const float4* hp = (const float4*)(h1 + (size_t)s * HIDDEN);
    float4 v = hp[lane];
    float* aptr = agg + (size_t)d * HIDDEN + lane * 4;
    atomicAdd(aptr + 0, v.x * coef);
    atomicAdd(aptr + 1, v.y * coef);
    atomicAdd(aptr + 2, v.z * coef);
    atomicAdd(aptr + 3, v.w * coef);
}

// Self-loop + bias + ReLU, in place on agg (becomes h2).
__global__ void k_self_relu(float* __restrict__ agg, const float* __restrict__ h1,
                            const float* __restrict__ invsq, const float* __restrict__ b1,
                            int n) {
    long long t = (long long)blockIdx.x * blockDim.x + threadIdx.x;
    if (t >= (long long)n * HIDDEN) return;
    int node = (int)(t >> 7);
    int c    = (int)(t & (HIDDEN - 1));
    float is = invsq[node];
    float v = agg[t] + h1[t] * is * is + b1[c];
    agg[t] = v > 0.f ? v : 0.f;
}

// ---------------------------------------------------------------------------
// GEMM2: h3[n,40] = h2[n,128] @ W2[128,40].  3 N-tiles, cols >= 40 masked.
__global__ void k_gemm2(const float* __restrict__ h2, const float* __restrict__ W2,
                        float* __restrict__ h3, int n) {
    int wave = (int)((blockIdx.x * blockDim.x + threadIdx.x) >> 5);
    int lane = threadIdx.x & 31;
    int m0 = wave * 16;
    if (m0 >= n) return;

    int row = lane & 15;
    int kh  = (lane >> 4) << 1;

    v8f acc[3];
#pragma unroll
    for (int t = 0; t < 3; ++t) acc[t] = (v8f){0.f,0.f,0.f,0.f,0.f,0.f,0.f,0.f};

    int arow = m0 + row; if (arow >= n) arow = n - 1;
    const float* ap = h2 + (size_t)arow * HIDDEN;

    for (int k = 0; k < HIDDEN; k += 4) {
        v2f a;
        a.x = ap[k + kh + 0];
        a.y = ap[k + kh + 1];
#pragma unroll
        for (int t = 0; t < 3; ++t) {
            int col = t * 16 + row;
            v2f b;
            b.x = (col < NCLS) ? W2[(size_t)(k + kh) * NCLS + col] : 0.f;
            b.y = (col < NCLS) ? W2[(size_t)(k + kh + 1) * NCLS + col] : 0.f;
            acc[t] = __builtin_amdgcn_wmma_f32_16x16x4_f32(
                false, a, false, b, (short)0, acc[t], false, false);
        }
    }

    int mo = (lane >> 4) * 8;
    int nc = lane & 15;
#pragma unroll
    for (int t = 0; t < 3; ++t) {
        int col = t * 16 + nc;
        if (col >= NCLS) continue;
#pragma unroll
        for (int r = 0; r < 8; ++r) {
            int mrow = m0 + mo + r;
            if (mrow < n) h3[(size_t)mrow * NCLS + col] = acc[t][r];
        }
    }
}

// Edge scatter layer 2: 40 channels; lane c handles c and (c+32 if c<8).
__global__ void k_edge2(const int* __restrict__ src, const int* __restrict__ dst,
                        const float* __restrict__ h3, const float* __restrict__ invsq,
                        float* __restrict__ agg, int e) {
    long long gid = (long long)blockIdx.x * blockDim.x + threadIdx.x;
    int edge = (int)(gid >> 5);
    int lane = (int)(gid & 31);
    if (edge >= e) return;
    int s = src[edge], d = dst[edge];
    float coef = invsq[s] * invsq[d];
    const float* hp = h3 + (size_t)s * NCLS;
    float* aptr = agg + (size_t)d * NCLS;
    atomicAdd(aptr + lane, hp[lane] * coef);
    if (lane < NCLS - 32)
        atomicAdd(aptr + lane + 32, hp[lane + 32] * coef);
}

// Final: out = log_softmax(agg2 + h3*invsq^2 + b2).  One wave per node.
__global__ void k_final(const float* __restrict__ agg2, const float* __restrict__ h3,
                        const float* __restrict__ invsq, const float* __restrict__ b2,
                        float* __restrict__ out, int n) {
    long long gid = (long long)blockIdx.x * blockDim.x + threadIdx.x;
    int node = (int)(gid >> 5);
    int lane = (int)(gid & 31);
    if (node >= n) return;
    float is = invsq[node];
    float is2 = is * is;
    const float* a = agg2 + (size_t)node * NCLS;
    const float* h = h3 + (size_t)node * NCLS;
    bool has2 = lane < (NCLS - 32);

    float x0 = a[lane] + h[lane] * is2 + b2[lane];
    float x1 = has2 ? (a[lane + 32] + h[lane + 32] * is2 + b2[lane + 32]) : 0.f;

    float m = has2 ? fmaxf(x0, x1) : x0;
#pragma unroll
    for (int off = 16; off > 0; off >>= 1)
        m = fmaxf(m, __shfl_xor(m, off, 32));

    float sum = expf(x0 - m) + (has2 ? expf(x1 - m) : 0.f);
#pragma unroll
    for (int off = 16; off > 0; off >>= 1)
        sum += __shfl_xor(sum, off, 32);

    float lse = m + logf(sum);
    out[(size_t)node * NCLS + lane] = x0 - lse;
    if (has2) out[(size_t)node * NCLS + lane + 32] = x1 - lse;
}

// ---------------------------------------------------------------------------
extern "C" void kernel_launch(void* const* d_in, const int* in_sizes, int n_in,
                              void* d_out, int out_size, void* d_ws, size_t ws_size,
                              hipStream_t stream) {
    const float* x  = (const float*)d_in[0];
    const int*   ei = (const int*)  d_in[1];
    const float* W1 = (const float*)d_in[2];
    const float* b1 = (const float*)d_in[3];
    const float* W2 = (const float*)d_in[4];
    const float* b2 = (const float*)d_in[5];
    float* out = (float*)d_out;

    int n = in_sizes[0] / NFEAT;       // 100000
    int e = in_sizes[1] / 2;           // 1600000
    const int* src = ei;               // edge_index[0]
    const int* dst = ei + e;           // edge_index[1]

    // Workspace layout (h3 & agg2 overlay the dead h1 region):
    auto align256 = [](size_t v) { return (v + 255) & ~(size_t)255; };
    char* ws = (char*)d_ws;
    size_t off_inv = 0;
    size_t off_h1  = align256((size_t)n * sizeof(float));
    size_t off_h2  = off_h1 + align256((size_t)n * HIDDEN * sizeof(float));
    float* invsq = (float*)(ws + off_inv);
    float* h1    = (float*)(ws + off_h1);
    float* h2    = (float*)(ws + off_h2);          // agg1 -> relu in place
    float* h3    = (float*)(ws + off_h1);          // reuse h1 region
    float* agg2  = (float*)(ws + off_h1 + align256((size_t)n * NCLS * sizeof(float)));

    // Zero accumulators (deterministic start each call).
    hipMemsetAsync(invsq, 0, (size_t)n * sizeof(float), stream);
    hipMemsetAsync(h2,    0, (size_t)n * HIDDEN * sizeof(float), stream);

    // Degree + normalization coefficients.
    k_degree <<<(e + 255) / 256, 256, 0, stream>>>(dst, invsq, e);
    k_invsqrt<<<(n + 255) / 256, 256, 0, stream>>>(invsq, n);

    // Layer 1.
    {
        int waves = (n + 15) / 16;
        long long threads = (long long)waves * 32;
        k_gemm1<<<(unsigned)((threads + 127) / 128), 128, 0, stream>>>(x, W1, h1, n);
    }
    {
        long long threads = (long long)e * 32;
        k_edge1<<<(unsigned)((threads + 255) / 256), 256, 0, stream>>>(src, dst, h1, invsq, h2, e);
    }
    {
        long long total = (long long)n * HIDDEN;
        k_self_relu<<<(unsigned)((total + 255) / 256), 256, 0, stream>>>(h2, h1, invsq, b1, n);
    }

    // Layer 2 (h1 region now dead -> h3/agg2 live there).
    hipMemsetAsync(agg2, 0, (size_t)n * NCLS * sizeof(float), stream);
    {
        int waves = (n + 15) / 16;
        long long threads = (long long)waves * 32;
        k_gemm2<<<(unsigned)((threads + 127) / 128), 128, 0, stream>>>(h2, W2, h3, n);
    }
    {
        long long threads = (long long)e * 32;
        k_edge2<<<(unsigned)((threads + 255) / 256), 256, 0, stream>>>(src, dst, h3, invsq, agg2, e);
    }
    {
        long long threads = (long long)n * 32;
        k_final<<<(unsigned)((threads + 255) / 256), 256, 0, stream>>>(agg2, h3, invsq, b2, out, n);
    }
}